// GATRegression_40939628265741
// MI455X (gfx1250) — compile-verified
//
#include <hip/hip_runtime.h>
#include <hip/hip_bf16.h>

typedef __attribute__((ext_vector_type(2))) float v2f;
typedef __attribute__((ext_vector_type(8))) float v8f;

#define IN_CH 128
#define HEADS 8
#define HID 16
#define NEG_SLOPE 0.2f

// ---- order-preserving float <-> uint key for atomicMax-based segment max ----
__device__ __forceinline__ unsigned int fkey(float f) {
    unsigned int u = __float_as_uint(f);
    return (u & 0x80000000u) ? ~u : (u | 0x80000000u);
}
__device__ __forceinline__ float funkey(unsigned int k) {
    unsigned int u = (k & 0x80000000u) ? (k & 0x7fffffffu) : ~k;
    return __uint_as_float(u);
}
__device__ __forceinline__ float lrelu(float x) { return x > 0.f ? x : NEG_SLOPE * x; }

// ---------------- init: zero accumulators, init max-keys, seed d_out=b2 ------
__global__ __launch_bounds__(256) void init_kernel(
    float* __restrict__ out1, unsigned int* __restrict__ emax1, float* __restrict__ denom1,
    unsigned int* __restrict__ emax2, float* __restrict__ denom2,
    float* __restrict__ dout, const float* __restrict__ b2, int N)
{
    int i = blockIdx.x * blockDim.x + threadIdx.x;
    if (i < N * 128) out1[i] = 0.f;
    if (i < N * 8) { emax1[i] = 0u; denom1[i] = 0.f; }
    if (i < N)     { emax2[i] = 0u; denom2[i] = 0.f; dout[i] = b2[0]; }
}

// ---------------- h1 = x @ W1 via V_WMMA_F32_16X16X4_F32 ---------------------
// one wave per 16x16 output tile; K=128 stepped by 4 (32 WMMA ops/tile)
__global__ __launch_bounds__(32) void gemm1_wmma(
    const float* __restrict__ x, const float* __restrict__ W,
    float* __restrict__ h1, int N)
{
    const int m0   = blockIdx.x * 16;
    const int n0   = blockIdx.y * 16;
    const int lane = threadIdx.x;          // 0..31 (wave32)
    const int r    = lane & 15;
    const int kb   = (lane >> 4) * 2;      // K sub-pair selector

    int arow = m0 + r; if (arow >= N) arow = N - 1;   // clamp (EXEC must stay all-1)
    const float* xrow = x + (size_t)arow * IN_CH;

    v8f c = {};
    for (int k = 0; k < IN_CH; k += 4) {
        v2f a, b;
        a.x = xrow[k + kb];
        a.y = xrow[k + kb + 1];
        b.x = W[(size_t)(k + kb)     * 128 + n0 + r];
        b.y = W[(size_t)(k + kb + 1) * 128 + n0 + r];
        c = __builtin_amdgcn_wmma_f32_16x16x4_f32(
                false, a, false, b, (short)0, c, false, false);
    }

    const int mbase = m0 + (lane >> 4) * 8;
    float* op = h1 + n0 + r;
#pragma unroll
    for (int v = 0; v < 8; ++v) {
        int row = mbase + v;
        if (row < N) op[(size_t)row * 128] = c[v];
    }
}

// ---------------- per-(node,head) attention dots -----------------------------
__global__ __launch_bounds__(256) void att_node1(
    const float* __restrict__ h1, const float* __restrict__ att_src,
    const float* __restrict__ att_dst,
    float* __restrict__ a_s, float* __restrict__ a_d, int N)
{
    int i = blockIdx.x * blockDim.x + threadIdx.x;      // node*8 + head
    if (i >= N * 8) return;
    int h = i & 7;
    const float* hp = h1 + (size_t)(i >> 3) * 128 + h * HID;
    const float* as = att_src + h * HID;
    const float* ad = att_dst + h * HID;
    float s = 0.f, d = 0.f;
#pragma unroll
    for (int c = 0; c < HID; ++c) { float v = hp[c]; s += v * as[c]; d += v * ad[c]; }
    a_s[i] = s; a_d[i] = d;
}

__device__ __forceinline__ void edge_nodes(const int* __restrict__ ei, int idx,
                                           int E, int& src, int& dst)
{
    if (idx < E) { src = ei[idx]; dst = ei[E + idx]; }
    else         { src = dst = idx - E; }   // self loops appended
}

// ---------------- layer-1 edge pass A: segment max ---------------------------
__global__ __launch_bounds__(256) void edge1_max(
    const int* __restrict__ ei, const float* __restrict__ a_s, const float* __restrict__ a_d,
    unsigned int* __restrict__ emax, int E, int Etot)
{
    int idx = blockIdx.x * blockDim.x + threadIdx.x;
    if (idx >= Etot) return;
    int src, dst; edge_nodes(ei, idx, E, src, dst);
#pragma unroll
    for (int h = 0; h < HEADS; ++h) {
        float e = lrelu(a_s[src * 8 + h] + a_d[dst * 8 + h]);
        atomicMax(&emax[dst * 8 + h], fkey(e));
    }
}

// ---------------- layer-1 edge pass B: segment sum of exp --------------------
__global__ __launch_bounds__(256) void edge1_sum(
    const int* __restrict__ ei, const float* __restrict__ a_s, const float* __restrict__ a_d,
    const unsigned int* __restrict__ emax, float* __restrict__ denom, int E, int Etot)
{
    int idx = blockIdx.x * blockDim.x + threadIdx.x;
    if (idx >= Etot) return;
    int src, dst; edge_nodes(ei, idx, E, src, dst);
#pragma unroll
    for (int h = 0; h < HEADS; ++h) {
        float e = lrelu(a_s[src * 8 + h] + a_d[dst * 8 + h]);
        float p = __expf(e - funkey(emax[dst * 8 + h]));
        atomicAdd(&denom[dst * 8 + h], p);
    }
}

// ---------------- layer-1 edge pass C: weighted scatter-add ------------------
// one wave per edge: lane handles 4 consecutive channels (head = lane/4)
__global__ __launch_bounds__(256) void edge1_aggr(
    const int* __restrict__ ei, const float* __restrict__ a_s, const float* __restrict__ a_d,
    const unsigned int* __restrict__ emax, const float* __restrict__ denom,
    const float* __restrict__ h1, float* __restrict__ out1, int E, int Etot)
{
    int widx = (blockIdx.x * blockDim.x + threadIdx.x) >> 5;   // edge id
    int lane = threadIdx.x & 31;
    if (widx >= Etot) return;
    int src, dst; edge_nodes(ei, widx, E, src, dst);

    int h = lane >> 2;                                         // head for my channels
    float e = lrelu(a_s[src * 8 + h] + a_d[dst * 8 + h]);
    float p = __expf(e - funkey(emax[dst * 8 + h]));
    float alpha = p / (denom[dst * 8 + h] + 1e-16f);

    int c0 = lane * 4;
    const float4 hv = *reinterpret_cast<const float4*>(h1 + (size_t)src * 128 + c0);
    float* o = out1 + (size_t)dst * 128 + c0;
    atomicAdd(o + 0, hv.x * alpha);
    atomicAdd(o + 1, hv.y * alpha);
    atomicAdd(o + 2, hv.z * alpha);
    atomicAdd(o + 3, hv.w * alpha);
}

// ---------------- node pass 2: ELU + 128->1 projection + attn scalars --------
// one wave per node; h2 never materialized
__global__ __launch_bounds__(256) void node2_kernel(
    const float* __restrict__ out1, const float* __restrict__ b1,
    const float* __restrict__ W2, const float* __restrict__ att_src2,
    const float* __restrict__ att_dst2,
    float* __restrict__ g2, float* __restrict__ a_s2, float* __restrict__ a_d2, int N)
{
    int n = (blockIdx.x * blockDim.x + threadIdx.x) >> 5;
    int lane = threadIdx.x & 31;
    if (n >= N) return;
    int c0 = lane * 4;
    float4 v  = *reinterpret_cast<const float4*>(out1 + (size_t)n * 128 + c0);
    float4 bb = *reinterpret_cast<const float4*>(b1 + c0);
    float4 w  = *reinterpret_cast<const float4*>(W2 + c0);
    float s = 0.f, t;
    t = v.x + bb.x; t = t > 0.f ? t : __expf(t) - 1.f; s += t * w.x;
    t = v.y + bb.y; t = t > 0.f ? t : __expf(t) - 1.f; s += t * w.y;
    t = v.z + bb.z; t = t > 0.f ? t : __expf(t) - 1.f; s += t * w.z;
    t = v.w + bb.w; t = t > 0.f ? t : __expf(t) - 1.f; s += t * w.w;
#pragma unroll
    for (int off = 16; off; off >>= 1) s += __shfl_xor(s, off, 32);
    if (lane == 0) { g2[n] = s; a_s2[n] = s * att_src2[0]; a_d2[n] = s * att_dst2[0]; }
}

// ---------------- layer-2 edge passes (1 head) -------------------------------
__global__ __launch_bounds__(256) void edge2_max(
    const int* __restrict__ ei, const float* __restrict__ a_s, const float* __restrict__ a_d,
    unsigned int* __restrict__ emax, int E, int Etot)
{
    int idx = blockIdx.x * blockDim.x + threadIdx.x;
    if (idx >= Etot) return;
    int src, dst; edge_nodes(ei, idx, E, src, dst);
    atomicMax(&emax[dst], fkey(lrelu(a_s[src] + a_d[dst])));
}

__global__ __launch_bounds__(256) void edge2_sum(
    const int* __restrict__ ei, const float* __restrict__ a_s, const float* __restrict__ a_d,
    const unsigned int* __restrict__ emax, float* __restrict__ denom, int E, int Etot)
{
    int idx = blockIdx.x * blockDim.x + threadIdx.x;
    if (idx >= Etot) return;
    int src, dst; edge_nodes(ei, idx, E, src, dst);
    float p = __expf(lrelu(a_s[src] + a_d[dst]) - funkey(emax[dst]));
    atomicAdd(&denom[dst], p);
}

__global__ __launch_bounds__(256) void edge2_aggr(
    const int* __restrict__ ei, const float* __restrict__ a_s, const float* __restrict__ a_d,
    const unsigned int* __restrict__ emax, const float* __restrict__ denom,
    const float* __restrict__ g2, float* __restrict__ dout, int E, int Etot)
{
    int idx = blockIdx.x * blockDim.x + threadIdx.x;
    if (idx >= Etot) return;
    int src, dst; edge_nodes(ei, idx, E, src, dst);
    float p = __expf(lrelu(a_s[src] + a_d[dst]) - funkey(emax[dst]));
    float alpha = p / (denom[dst] + 1e-16f);
    atomicAdd(&dout[dst], g2[src] * alpha);
}

extern "C" void kernel_launch(void* const* d_in, const int* in_sizes, int n_in,
                              void* d_out, int out_size, void* d_ws, size_t ws_size,
                              hipStream_t stream)
{
    const float* x        = (const float*)d_in[0];
    const int*   ei       = (const int*)  d_in[1];
    const float* W1       = (const float*)d_in[2];
    const float* att_src1 = (const float*)d_in[3];
    const float* att_dst1 = (const float*)d_in[4];
    const float* b1       = (const float*)d_in[5];
    const float* W2       = (const float*)d_in[6];
    const float* att_src2 = (const float*)d_in[7];
    const float* att_dst2 = (const float*)d_in[8];
    const float* b2       = (const float*)d_in[9];
    float* dout = (float*)d_out;

    const int N    = in_sizes[0] / IN_CH;   // 50000
    const int E    = in_sizes[1] / 2;       // 1600000
    const int Etot = E + N;                 // + self loops

    // workspace carve-up (floats / uints, 4B each)
    float* ws = (float*)d_ws;
    float*        h1     = ws;                              // N*128
    float*        out1   = h1     + (size_t)N * 128;        // N*128
    float*        a_s1   = out1   + (size_t)N * 128;        // N*8
    float*        a_d1   = a_s1   + (size_t)N * 8;          // N*8
    unsigned int* emax1  = (unsigned int*)(a_d1 + (size_t)N * 8);   // N*8
    float*        denom1 = (float*)(emax1 + (size_t)N * 8); // N*8
    float*        g2     = denom1 + (size_t)N * 8;          // N
    float*        a_s2   = g2     + N;                      // N
    float*        a_d2   = a_s2   + N;                      // N
    unsigned int* emax2  = (unsigned int*)(a_d2 + N);       // N
    float*        denom2 = (float*)(emax2 + N);             // N

    // 1) init accumulators + seed output with b2
    {
        int th = N * 128;
        init_kernel<<<(th + 255) / 256, 256, 0, stream>>>(out1, emax1, denom1,
                                                          emax2, denom2, dout, b2, N);
    }
    // 2) h1 = x @ W1  (WMMA f32 16x16x4)
    {
        dim3 grid((N + 15) / 16, (HEADS * HID) / 16);
        gemm1_wmma<<<grid, 32, 0, stream>>>(x, W1, h1, N);
    }
    // 3) per-(node,head) attention dots
    att_node1<<<(N * 8 + 255) / 256, 256, 0, stream>>>(h1, att_src1, att_dst1,
                                                       a_s1, a_d1, N);
    // 4) layer-1 softmax over incoming edges + aggregation
    edge1_max <<<(Etot + 255) / 256, 256, 0, stream>>>(ei, a_s1, a_d1, emax1, E, Etot);
    edge1_sum <<<(Etot + 255) / 256, 256, 0, stream>>>(ei, a_s1, a_d1, emax1, denom1, E, Etot);
    {
        long long th = (long long)Etot * 32;
        edge1_aggr<<<(int)((th + 255) / 256), 256, 0, stream>>>(ei, a_s1, a_d1, emax1,
                                                                denom1, h1, out1, E, Etot);
    }
    // 5) ELU + projection to scalar + layer-2 attention scalars
    {
        long long th = (long long)N * 32;
        node2_kernel<<<(int)((th + 255) / 256), 256, 0, stream>>>(out1, b1, W2,
                                                                  att_src2, att_dst2,
                                                                  g2, a_s2, a_d2, N);
    }
    // 6) layer-2 softmax + aggregation into d_out (pre-seeded with b2)
    edge2_max <<<(Etot + 255) / 256, 256, 0, stream>>>(ei, a_s2, a_d2, emax2, E, Etot);
    edge2_sum <<<(Etot + 255) / 256, 256, 0, stream>>>(ei, a_s2, a_d2, emax2, denom2, E, Etot);
    edge2_aggr<<<(Etot + 255) / 256, 256, 0, stream>>>(ei, a_s2, a_d2, emax2, denom2,
                                                       g2, dout, E, Etot);
}